// CGDNBlock_9964324127083
// MI455X (gfx1250) — compile-verified
//
#include <hip/hip_runtime.h>
#include <math.h>

#define DD 128          // hidden channels
#define NH 4            // heads
#define NEG_SLOPE 0.2f
#define LN_EPS 1e-5f

typedef float v2f __attribute__((ext_vector_type(2)));
typedef float v8f __attribute__((ext_vector_type(8)));

// ---- order-preserving float <-> uint key for atomicMax-based segment max ----
__device__ __forceinline__ unsigned enc_f32(float f) {
    unsigned u = __float_as_uint(f);
    return (u & 0x80000000u) ? ~u : (u | 0x80000000u);
}
__device__ __forceinline__ float dec_f32(unsigned k) {
    return (k & 0x80000000u) ? __uint_as_float(k & 0x7FFFFFFFu)
                             : __uint_as_float(~k);
}

// ============================================================================
// Kernel 1: degree + loop-attr sums (atomic scatter over edges)
// ============================================================================
__global__ __launch_bounds__(256) void k_deg_loop(const int* __restrict__ ei,
                                                  const float* __restrict__ eattr,
                                                  float* __restrict__ deg,
                                                  float* __restrict__ loop_sum,
                                                  int E_) {
    int t = blockIdx.x * blockDim.x + threadIdx.x;
    if (t >= E_) return;
    int d = ei[E_ + t];                       // dst row of edge_index
    atomicAdd(deg + d, 1.0f);
    float4 ea = *(const float4*)(eattr + (size_t)t * 4);
    atomicAdd(loop_sum + d * 4 + 0, ea.x);
    atomicAdd(loop_sum + d * 4 + 1, ea.y);
    atomicAdd(loop_sum + d * 4 + 2, ea.z);
    atomicAdd(loop_sum + d * 4 + 3, ea.w);
}

// ============================================================================
// Kernel 2: loop_attr = loop_sum / max(deg, 1)
// ============================================================================
__global__ __launch_bounds__(256) void k_loop_div(float* __restrict__ loop_attr,
                                                  const float* __restrict__ deg,
                                                  int N_) {
    int t = blockIdx.x * blockDim.x + threadIdx.x;
    if (t >= N_ * 4) return;
    loop_attr[t] = loop_attr[t] / fmaxf(deg[t >> 2], 1.0f);
}

// ============================================================================
// Kernel 3: x = h @ W + b via V_WMMA_F32_16X16X4_F32 (fp32 tensor path).
// One wave per 16-row tile; h tile staged in LDS (stride 132 to dodge bank
// conflicts), W streamed from global (64 KB, L2/WGP$-resident).
// A-frag (16x4 f32): lane m=lane&15 holds row M=m; K = vgpr + 2*(lane>=16).
// B-frag (4x16 f32): lane holds col N=m; K = vgpr + 2*(lane>=16).
// C/D:               vgpr j holds row M=j+8*(lane>=16), col N=m.
// ============================================================================
__global__ __launch_bounds__(32) void k_proj_wmma(const float* __restrict__ hin,
                                                  const float* __restrict__ Wl,
                                                  const float* __restrict__ bl,
                                                  const float* __restrict__ Wr,
                                                  const float* __restrict__ br,
                                                  float* __restrict__ xl,
                                                  float* __restrict__ xr,
                                                  int N_) {
    const float* W;
    const float* b;
    float* out;
    if (blockIdx.y == 0) { W = Wl; b = bl; out = xl; }
    else                 { W = Wr; b = br; out = xr; }

    const int r0 = blockIdx.x * 16;
    const int lane = threadIdx.x;

    __shared__ float sA[16 * 132];
    // stage 16x128 h tile, coalesced float4 loads
    #pragma unroll
    for (int i = 0; i < 16; ++i) {
        int f4 = i * 32 + lane;
        int row = f4 >> 5;
        int c4 = (f4 & 31) * 4;
        int grow = min(r0 + row, N_ - 1);
        *(float4*)(sA + row * 132 + c4) =
            *(const float4*)(hin + (size_t)grow * DD + c4);
    }
    __syncthreads();

    const int half = lane >> 4;   // upper lane half
    const int m    = lane & 15;

    v8f acc[8] = {};
    for (int k = 0; k < DD; k += 4) {
        v2f a;
        const float* ap = sA + m * 132 + k + 2 * half;
        a.x = ap[0];
        a.y = ap[1];
        #pragma unroll
        for (int n = 0; n < 8; ++n) {
            v2f bf;
            const float* bp = W + (size_t)(k + 2 * half) * DD + n * 16 + m;
            bf.x = bp[0];
            bf.y = bp[DD];
            acc[n] = __builtin_amdgcn_wmma_f32_16x16x4_f32(
                false, a, false, bf, (short)0, acc[n], false, false);
        }
    }

    #pragma unroll
    for (int n = 0; n < 8; ++n) {
        int col = n * 16 + m;
        float bc = b[col];
        #pragma unroll
        for (int j = 0; j < 8; ++j) {
            int row = r0 + j + 8 * half;
            if (row < N_) out[(size_t)row * DD + col] = acc[n][j] + bc;
        }
    }
}

// ============================================================================
// Kernel 4: per-edge attention logits. One wave per edge; lane owns 4
// consecutive channels (head = lane>>3). e_proj (K=4) fused. Segment max via
// atomicMax on order-preserving uint keys.
// ============================================================================
__global__ __launch_bounds__(256) void k_edge_alpha(const int* __restrict__ ei,
                                                    const float* __restrict__ eattr,
                                                    const float* __restrict__ loop_attr,
                                                    const float* __restrict__ xl,
                                                    const float* __restrict__ xr,
                                                    const float* __restrict__ We,
                                                    const float* __restrict__ att,
                                                    float* __restrict__ alphas,
                                                    unsigned* __restrict__ amax,
                                                    int E_, int N_) {
    int wid = blockIdx.x * (blockDim.x >> 5) + (threadIdx.x >> 5);
    int lane = threadIdx.x & 31;
    int EN = E_ + N_;
    if (wid >= EN) return;

    int s, d;
    const float* eap;
    if (wid < E_) { s = ei[wid]; d = ei[E_ + wid]; eap = eattr + (size_t)wid * 4; }
    else          { s = wid - E_; d = s;           eap = loop_attr + (size_t)s * 4; }
    float4 ea = *(const float4*)eap;

    int c0 = lane * 4;
    float4 a4 = *(const float4*)(xl + (size_t)s * DD + c0);
    float4 r4 = *(const float4*)(xr + (size_t)d * DD + c0);
    float4 w0 = *(const float4*)(We + 0 * DD + c0);
    float4 w1 = *(const float4*)(We + 1 * DD + c0);
    float4 w2 = *(const float4*)(We + 2 * DD + c0);
    float4 w3 = *(const float4*)(We + 3 * DD + c0);

    float4 v;
    v.x = a4.x + r4.x + ea.x * w0.x + ea.y * w1.x + ea.z * w2.x + ea.w * w3.x;
    v.y = a4.y + r4.y + ea.x * w0.y + ea.y * w1.y + ea.z * w2.y + ea.w * w3.y;
    v.z = a4.z + r4.z + ea.x * w0.z + ea.y * w1.z + ea.z * w2.z + ea.w * w3.z;
    v.w = a4.w + r4.w + ea.x * w0.w + ea.y * w1.w + ea.z * w2.w + ea.w * w3.w;
    v.x = v.x > 0.f ? v.x : NEG_SLOPE * v.x;
    v.y = v.y > 0.f ? v.y : NEG_SLOPE * v.y;
    v.z = v.z > 0.f ? v.z : NEG_SLOPE * v.z;
    v.w = v.w > 0.f ? v.w : NEG_SLOPE * v.w;

    // att[(1,H,C)] flat: head*32 + (lane&7)*4 == lane*4
    float4 at = *(const float4*)(att + c0);
    float p = v.x * at.x + v.y * at.y + v.z * at.z + v.w * at.w;
    p += __shfl_xor(p, 1, 32);
    p += __shfl_xor(p, 2, 32);
    p += __shfl_xor(p, 4, 32);     // reduced over the 8 lanes of this head

    if ((lane & 7) == 0) {
        int head = lane >> 3;
        alphas[(size_t)wid * NH + head] = p;
        atomicMax(amax + (size_t)d * NH + head, enc_f32(p));
    }
}

// ============================================================================
// Kernel 5: ex = exp(alpha - max); denom += ex. One thread per (edge, head).
// ============================================================================
__global__ __launch_bounds__(256) void k_edge_exp(const int* __restrict__ ei,
                                                  float* __restrict__ alphas,
                                                  const unsigned* __restrict__ amax,
                                                  float* __restrict__ denom,
                                                  int E_, int N_) {
    int t = blockIdx.x * blockDim.x + threadIdx.x;
    int EN4 = (E_ + N_) * NH;
    if (t >= EN4) return;
    int e = t >> 2;
    int hd = t & 3;
    int d = (e < E_) ? ei[E_ + e] : (e - E_);
    float ex = __expf(alphas[t] - dec_f32(amax[d * NH + hd]));
    alphas[t] = ex;
    atomicAdd(denom + d * NH + hd, ex);
}

// ============================================================================
// Kernel 6: weighted scatter-aggregation. Wave per edge; x_l gather is
// L2-resident (51 MB < 192 MB L2); scatter via global_atomic_add_f32.
// ============================================================================
__global__ __launch_bounds__(256) void k_edge_agg(const int* __restrict__ ei,
                                                  const float* __restrict__ xl,
                                                  const float* __restrict__ alphas,
                                                  const float* __restrict__ denom,
                                                  float* __restrict__ oacc,
                                                  int E_, int N_) {
    int wid = blockIdx.x * (blockDim.x >> 5) + (threadIdx.x >> 5);
    int lane = threadIdx.x & 31;
    int EN = E_ + N_;
    if (wid >= EN) return;

    int s, d;
    if (wid < E_) { s = ei[wid]; d = ei[E_ + wid]; }
    else          { s = wid - E_; d = s; }

    int head = lane >> 3;
    float w = alphas[(size_t)wid * NH + head] / denom[(size_t)d * NH + head];

    int c0 = lane * 4;
    float4 a4 = *(const float4*)(xl + (size_t)s * DD + c0);
    float* op = oacc + (size_t)d * DD + c0;
    atomicAdd(op + 0, a4.x * w);
    atomicAdd(op + 1, a4.y * w);
    atomicAdd(op + 2, a4.z * w);
    atomicAdd(op + 3, a4.w * w);
}

// ============================================================================
// Kernel 7: bias + LayerNorm + FiLM + exact GELU + residual. Wave per node.
// ============================================================================
__global__ __launch_bounds__(256) void k_final(const float* __restrict__ oacc,
                                               const float* __restrict__ bias,
                                               const float* __restrict__ lnw,
                                               const float* __restrict__ lnb,
                                               const float* __restrict__ gamma,
                                               const float* __restrict__ beta,
                                               const float* __restrict__ hin,
                                               float* __restrict__ y,
                                               int N_) {
    int wid = blockIdx.x * (blockDim.x >> 5) + (threadIdx.x >> 5);
    int lane = threadIdx.x & 31;
    if (wid >= N_) return;
    int c0 = lane * 4;

    float4 o = *(const float4*)(oacc + (size_t)wid * DD + c0);
    float4 bs = *(const float4*)(bias + c0);
    o.x += bs.x; o.y += bs.y; o.z += bs.z; o.w += bs.w;

    float s = o.x + o.y + o.z + o.w;
    float sq = o.x * o.x + o.y * o.y + o.z * o.z + o.w * o.w;
    #pragma unroll
    for (int off = 1; off < 32; off <<= 1) {
        s += __shfl_xor(s, off, 32);
        sq += __shfl_xor(sq, off, 32);
    }
    float mu = s * (1.0f / DD);
    float var = sq * (1.0f / DD) - mu * mu;
    float rstd = rsqrtf(var + LN_EPS);

    float4 w4 = *(const float4*)(lnw + c0);
    float4 b4 = *(const float4*)(lnb + c0);
    float4 g4 = *(const float4*)(gamma + (size_t)wid * DD + c0);
    float4 be4 = *(const float4*)(beta + (size_t)wid * DD + c0);
    float4 h4 = *(const float4*)(hin + (size_t)wid * DD + c0);

    float4 r;
    {
        float hn = (o.x - mu) * rstd * w4.x + b4.x;
        float t = g4.x * hn + be4.x;
        r.x = 0.5f * t * (1.0f + erff(t * 0.70710678118654752f)) + h4.x;
    }
    {
        float hn = (o.y - mu) * rstd * w4.y + b4.y;
        float t = g4.y * hn + be4.y;
        r.y = 0.5f * t * (1.0f + erff(t * 0.70710678118654752f)) + h4.y;
    }
    {
        float hn = (o.z - mu) * rstd * w4.z + b4.z;
        float t = g4.z * hn + be4.z;
        r.z = 0.5f * t * (1.0f + erff(t * 0.70710678118654752f)) + h4.z;
    }
    {
        float hn = (o.w - mu) * rstd * w4.w + b4.w;
        float t = g4.w * hn + be4.w;
        r.w = 0.5f * t * (1.0f + erff(t * 0.70710678118654752f)) + h4.w;
    }
    *(float4*)(y + (size_t)wid * DD + c0) = r;
}

// ============================================================================
extern "C" void kernel_launch(void* const* d_in, const int* in_sizes, int n_in,
                              void* d_out, int out_size, void* d_ws, size_t ws_size,
                              hipStream_t stream) {
    const float* h     = (const float*)d_in[0];
    const int*   ei    = (const int*)d_in[1];
    const float* eattr = (const float*)d_in[2];
    const float* gamma = (const float*)d_in[3];
    const float* beta  = (const float*)d_in[4];
    const float* Wl    = (const float*)d_in[5];
    const float* bl    = (const float*)d_in[6];
    const float* Wr    = (const float*)d_in[7];
    const float* br    = (const float*)d_in[8];
    const float* We    = (const float*)d_in[9];
    const float* att   = (const float*)d_in[10];
    const float* bias  = (const float*)d_in[11];
    const float* lnw   = (const float*)d_in[12];
    const float* lnb   = (const float*)d_in[13];

    const int N_ = in_sizes[0] / DD;
    const int E_ = in_sizes[1] / 2;
    const int EN = E_ + N_;

    // workspace carve-up (x_l + x_r = 102 MB -> L2-resident gather targets)
    char* ws = (char*)d_ws;
    size_t off = 0;
    auto carve = [&](size_t bytes) -> char* {
        char* p = ws + off;
        off += (bytes + 255) & ~(size_t)255;
        return p;
    };
    float*    xl     = (float*)carve((size_t)N_ * DD * 4);
    float*    xr     = (float*)carve((size_t)N_ * DD * 4);
    char*     zbeg   = ws + off;                       // zero-init region start
    float*    oacc   = (float*)carve((size_t)N_ * DD * 4);
    float*    deg    = (float*)carve((size_t)N_ * 4);
    float*    lattr  = (float*)carve((size_t)N_ * 4 * 4);
    unsigned* amax   = (unsigned*)carve((size_t)N_ * NH * 4);
    float*    denom  = (float*)carve((size_t)N_ * NH * 4);
    size_t    zlen   = (size_t)((ws + off) - zbeg);
    float*    alphas = (float*)carve((size_t)EN * NH * 4); // written before read

    hipMemsetAsync(zbeg, 0, zlen, stream);  // oacc/deg/lattr/amax/denom = 0
                                            // (amax key 0 == below all floats)

    k_deg_loop<<<(E_ + 255) / 256, 256, 0, stream>>>(ei, eattr, deg, lattr, E_);
    k_loop_div<<<(N_ * 4 + 255) / 256, 256, 0, stream>>>(lattr, deg, N_);

    dim3 gproj((N_ + 15) / 16, 2);
    k_proj_wmma<<<gproj, 32, 0, stream>>>(h, Wl, bl, Wr, br, xl, xr, N_);

    int wblocks = (EN + 7) / 8;   // 8 waves (edges) per 256-thread block
    k_edge_alpha<<<wblocks, 256, 0, stream>>>(ei, eattr, lattr, xl, xr, We, att,
                                              alphas, amax, E_, N_);
    k_edge_exp<<<(EN * NH + 255) / 256, 256, 0, stream>>>(ei, alphas, amax,
                                                          denom, E_, N_);
    k_edge_agg<<<wblocks, 256, 0, stream>>>(ei, xl, alphas, denom, oacc, E_, N_);
    k_final<<<(N_ + 7) / 8, 256, 0, stream>>>(oacc, bias, lnw, lnb, gamma, beta,
                                              h, (float*)d_out, N_);
}